// S_GC_att_A_16527034155345
// MI455X (gfx1250) — compile-verified
//
#include <hip/hip_runtime.h>
#include <hip/hip_bf16.h>

typedef __attribute__((ext_vector_type(2))) float v2f;
typedef __attribute__((ext_vector_type(8))) float v8f;
typedef int i32x4 __attribute__((vector_size(16)));

#define S_KER   3
#define NUM_ATT 4
#define K_TOTAL 7          // 3 + 4
#define C_IN    256
#define C_OUT   256
#define T_DIM   512
#define V_DIM   22
#define N_DIM   8

#define C_BLK   16                     // out channels per block
#define T_BLK   8                      // timesteps per block
#define P_DIM   (T_BLK * V_DIM)        // 176 = 11 WMMA N-tiles of 16
#define M_DIM   (K_TOTAL * C_BLK)      // 112 = 7 WMMA M-tiles of 16
#define KC      32                     // c_in chunk per LDS stage
#define NCHUNK  (C_IN / KC)            // 8
#define WS_PITCH 36                    // 144 B rows -> 16B-aligned b128 LDS writes
#define XS_PITCH 34
#define YS_PITCH 178
#define KV_PAD  24                     // v padded 22 -> 24 per k
#define AB_ROWS (K_TOTAL * KV_PAD)     // 168
#define AB_PITCH 32
#define NTHREADS 352                   // 11 wave32

#define WS_SIZE (M_DIM * WS_PITCH)               // 4032 floats per buffer
#define XS_SIZE (P_DIM * XS_PITCH)               // 5984 floats per buffer

// LDS layout (floats): double-buffered W/X tiles + Y + adjacency + bias
#define OFF_WS  0                                // two WS buffers back to back
#define OFF_XS  (OFF_WS + 2 * WS_SIZE)           // 8064 ; two XS buffers back to back
#define OFF_YS  (OFF_XS + 2 * XS_SIZE)           // 20032
#define OFF_AB  (OFF_YS + M_DIM * YS_PITCH)      // 39968
#define OFF_BS  (OFF_AB + AB_ROWS * AB_PITCH)    // 45344
#define LDS_FLOATS (OFF_BS + M_DIM)              // 45456 floats = 181,824 B (< 320KB WGP)

// ---- CDNA5 async global->LDS path (ASYNCcnt-tracked), with safe fallback ----
#if __has_builtin(__builtin_amdgcn_global_load_async_to_lds_b32) && \
    __has_builtin(__builtin_amdgcn_global_load_async_to_lds_b128)
#define ASYNC_LDS 1
#else
#define ASYNC_LDS 0
#endif

// Pointee-typed, address-space-qualified casts per clang's builtin signatures:
//   b32 : (int AS1*, int AS3*, Imm, Imm)
//   b128: (i32x4 AS1*, i32x4 AS3*, Imm, Imm)
#define GAS1_I32(p)  ((__attribute__((address_space(1))) int*)(p))
#define LAS3_I32(p)  ((__attribute__((address_space(3))) int*)(p))
#define GAS1_I128(p) ((__attribute__((address_space(1))) i32x4*)(p))
#define LAS3_I128(p) ((__attribute__((address_space(3))) i32x4*)(p))

__device__ __forceinline__ void async_copy_b32(const float* g, float* l) {
#if ASYNC_LDS
    __builtin_amdgcn_global_load_async_to_lds_b32(GAS1_I32(g), LAS3_I32(l), 0, 0);
#else
    *l = *g;
#endif
}

__device__ __forceinline__ void async_copy_b128(const float* g, float* l) {
#if ASYNC_LDS
    __builtin_amdgcn_global_load_async_to_lds_b128(GAS1_I128(g), LAS3_I128(l), 0, 0);
#else
    l[0] = g[0]; l[1] = g[1]; l[2] = g[2]; l[3] = g[3];
#endif
}

__device__ __forceinline__ void wait_async_lds() {
#if ASYNC_LDS
#if __has_builtin(__builtin_amdgcn_s_wait_asynccnt)
    __builtin_amdgcn_s_wait_asynccnt(0);
#else
    asm volatile("s_wait_asynccnt 0x0" ::: "memory");
#endif
#endif
}

// Issue the async staging of one K-chunk (W tile via b128, X tile transposed via b32 scatter)
__device__ __forceinline__ void issue_chunk(float* WsD, float* XsD,
                                            const float* __restrict__ W,
                                            const float* __restrict__ x,
                                            int cc, int c0, long xbase, int tid)
{
    // W tile: rows m = k*16+ci -> W[k*256 + c0 + ci][cc .. cc+31], 4 floats/lane
    for (int e = tid; e < M_DIM * (KC / 4); e += NTHREADS) {
        int ol = e >> 3, q = (e & 7) * 4;
        int k = ol >> 4, ci = ol & 15;
        async_copy_b128(&W[(k * C_OUT + c0 + ci) * C_IN + cc + q],
                        &WsD[ol * WS_PITCH + q]);
    }
    // X tile, transposed into Xs[p][cr]: per-lane LDS scatter address
    for (int e = tid; e < KC * P_DIM; e += NTHREADS) {
        int cr = e / P_DIM, p = e - cr * P_DIM;
        async_copy_b32(&x[xbase + (long)(cc + cr) * (T_DIM * V_DIM) + p],
                       &XsD[p * XS_PITCH + cr]);
    }
}

__global__ void __launch_bounds__(NTHREADS)
fused_stgcn_kernel(const float* __restrict__ x,     // (8,256,512,22)
                   const float* __restrict__ A,     // (3,22,22)
                   const float* __restrict__ attA,  // (8,4,22,22)
                   const float* __restrict__ W,     // (1792,256)
                   const float* __restrict__ b,     // (1792)
                   float* __restrict__ out)         // (8,256,512,22)
{
    extern __shared__ float lds[];
    float* WsBase = lds + OFF_WS;   // buffer i at WsBase + i*WS_SIZE
    float* XsBase = lds + OFF_XS;   // buffer i at XsBase + i*XS_SIZE
    float* Ys = lds + OFF_YS;
    float* Ab = lds + OFF_AB;
    float* bs = lds + OFF_BS;

    const int tid  = threadIdx.x;
    const int wave = tid >> 5;
    const int lane = tid & 31;
    const int half = lane >> 4;   // 0: lanes 0-15, 1: lanes 16-31
    const int row  = lane & 15;

    const int c0 = blockIdx.x * C_BLK;
    const int t0 = blockIdx.y * T_BLK;
    const int n  = blockIdx.z;

    const long xbase = ((long)n * C_IN) * (T_DIM * V_DIM) + (long)t0 * V_DIM;

    // Kick off chunk 0 staging immediately (overlaps with Ab/bias build below)
    issue_chunk(WsBase, XsBase, W, x, 0, c0, xbase, tid);

    // ---- Build combined, padded adjacency Ab[(k*24+v)][w] and bias slice ----
    for (int e = tid; e < AB_ROWS * AB_PITCH; e += NTHREADS) {
        int kvp = e >> 5;            // AB_PITCH == 32
        int w   = e & 31;
        int k   = kvp / KV_PAD;
        int v   = kvp - k * KV_PAD;
        float val = 0.0f;
        if (v < V_DIM && w < V_DIM) {
            if (k < S_KER) val = A[(k * V_DIM + v) * V_DIM + w];
            else           val = attA[((n * NUM_ATT + (k - S_KER)) * V_DIM + v) * V_DIM + w];
        }
        Ab[kvp * AB_PITCH + w] = val;
    }
    for (int e = tid; e < M_DIM; e += NTHREADS) {
        int k = e >> 4, ci = e & 15;
        bs[e] = b[k * C_OUT + c0 + ci];
    }

    wait_async_lds();
    __syncthreads();

    // ---- Stage A: Y[112][176] = W_blk @ X_blk (fp32 WMMA), double-buffered ----
    v8f acc[K_TOTAL] = {};
    const int ni = wave;             // each of the 11 waves owns one N-tile of 16 pixels

    for (int ch = 0; ch < NCHUNK; ++ch) {
        const int cur = ch & 1;
        if (ch + 1 < NCHUNK)   // prefetch next chunk into the other buffer (overlaps WMMA)
            issue_chunk(WsBase + (cur ^ 1) * WS_SIZE, XsBase + (cur ^ 1) * XS_SIZE,
                        W, x, (ch + 1) * KC, c0, xbase, tid);

        const float* Ws = WsBase + cur * WS_SIZE;
        const float* Xs = XsBase + cur * XS_SIZE;
        #pragma unroll
        for (int s = 0; s < KC / 4; ++s) {
            const int kc = s * 4 + 2 * half;   // VGPR0 holds K=0/K=2, VGPR1 K=1/K=3
            const float* xr = &Xs[(ni * 16 + row) * XS_PITCH + kc];
            v2f bf; bf.x = xr[0]; bf.y = xr[1];
            v2f af[K_TOTAL];
            #pragma unroll
            for (int mi = 0; mi < K_TOTAL; ++mi) {
                const float* wr = &Ws[(mi * 16 + row) * WS_PITCH + kc];
                af[mi].x = wr[0]; af[mi].y = wr[1];
            }
            #pragma unroll
            for (int mi = 0; mi < K_TOTAL; ++mi) {
                acc[mi] = __builtin_amdgcn_wmma_f32_16x16x4_f32(
                    false, af[mi], false, bf, (short)0, acc[mi], false, false);
            }
        }

        wait_async_lds();   // next chunk's async copies done (overlapped with WMMA above)
        __syncthreads();    // all waves finished reading cur buffers / writes visible
    }

    // ---- Spill Y (+bias) to LDS per the C/D layout (VGPR r: M=r / M=8+r) ----
    #pragma unroll
    for (int mi = 0; mi < K_TOTAL; ++mi) {
        #pragma unroll
        for (int r = 0; r < 8; ++r) {
            int M = mi * 16 + half * 8 + r;
            int N = ni * 16 + row;
            Ys[M * YS_PITCH + N] = acc[mi][r] + bs[M];
        }
    }
    __syncthreads();

    // ---- Stage B: per timestep, out[16c][22w] = Yb[16c][168kv] @ Ab[168][32] ----
    if (wave < T_BLK) {               // wave-uniform branch: EXEC stays all-1s for WMMA
        const int tl = wave;
        v8f o0 = {}, o1 = {};
        #pragma unroll 6
        for (int s = 0; s < AB_ROWS / 4; ++s) {
            const int k  = s / 6;                 // 6 K-steps per k (24/4)
            const int e0 = 4 * s + 2 * half;      // kv' index for this lane-half
            const int v0 = e0 - k * KV_PAD;
            const int v1 = v0 + 1;
            const float* yr = &Ys[(k * 16 + row) * YS_PITCH + tl * V_DIM];
            float a0 = yr[v0 < V_DIM ? v0 : 0] * (v0 < V_DIM ? 1.0f : 0.0f);
            float a1 = yr[v1 < V_DIM ? v1 : 0] * (v1 < V_DIM ? 1.0f : 0.0f);
            v2f af; af.x = a0; af.y = a1;
            const float* br0 = &Ab[e0 * AB_PITCH + row];
            const float* br1 = br0 + AB_PITCH;
            v2f bf0; bf0.x = br0[0];  bf0.y = br1[0];
            v2f bf1; bf1.x = br0[16]; bf1.y = br1[16];
            o0 = __builtin_amdgcn_wmma_f32_16x16x4_f32(false, af, false, bf0, (short)0, o0, false, false);
            o1 = __builtin_amdgcn_wmma_f32_16x16x4_f32(false, af, false, bf1, (short)0, o1, false, false);
        }
        const long obase = (((long)n * C_OUT + c0) * T_DIM + (t0 + tl)) * V_DIM;
        #pragma unroll
        for (int r = 0; r < 8; ++r) {
            const int  cl = half * 8 + r;
            const long cb = obase + (long)cl * (T_DIM * V_DIM);
            out[cb + row] = o0[r];
            if (16 + row < V_DIM) out[cb + 16 + row] = o1[r];
        }
    }
}

extern "C" void kernel_launch(void* const* d_in, const int* in_sizes, int n_in,
                              void* d_out, int out_size, void* d_ws, size_t ws_size,
                              hipStream_t stream) {
    const float* x    = (const float*)d_in[0];
    const float* A    = (const float*)d_in[1];
    const float* attA = (const float*)d_in[2];
    const float* W    = (const float*)d_in[3];
    const float* b    = (const float*)d_in[4];
    float*       out  = (float*)d_out;

    dim3 grid(C_OUT / C_BLK, T_DIM / T_BLK, N_DIM);   // (16, 64, 8) = 8192 blocks
    dim3 block(NTHREADS);                              // 11 wave32 = 11 N-tiles
    size_t shmem = (size_t)LDS_FLOATS * sizeof(float); // ~177.6 KB dynamic LDS
    fused_stgcn_kernel<<<grid, block, shmem, stream>>>(x, A, attA, W, b, out);
}